// Net_60249801228589
// MI455X (gfx1250) — compile-verified
//
#include <hip/hip_runtime.h>
#include <math.h>

typedef float v2f __attribute__((ext_vector_type(2)));
typedef float v4f __attribute__((ext_vector_type(4)));
typedef float v8f __attribute__((ext_vector_type(8)));

#define D      16384   // 128*128 elements per pattern
#define NPAT   16384   // number of patterns
#define EPSF   1e-8f
#define NSPLIT 32
#define ROWS_PER_SPLIT (NPAT / NSPLIT)   // 512

// ---------------------------------------------------------------------------
// Kernel 1: fused dots[n] = P[n]·x  and  sqnorms[n] = ||P[n]||^2
// 256 threads = 8 waves; each wave owns 16 rows and drives two independent
// V_WMMA_F32_16X16X4_F32 accumulation chains (x broadcast across B columns),
// plus 4 independent FMAC chains for the squared norms (no serial VALU stalls).
// P loads are non-temporal (1 GiB stream, consumed within the unroll window).
// ---------------------------------------------------------------------------
__global__ __launch_bounds__(256) void k_dots_norms(
    const float* __restrict__ P, const float* __restrict__ x,
    float* __restrict__ dots, float* __restrict__ sqnorms)
{
  __shared__ float xs[D];   // 64 KiB: stage x once per block (reused 8x)
  {
    const float4* x4  = (const float4*)x;
    float4*       xs4 = (float4*)xs;
    for (int i = threadIdx.x; i < D / 4; i += 256) xs4[i] = x4[i];
  }
  __syncthreads();

  const int lane    = threadIdx.x & 31;
  const int wave    = threadIdx.x >> 5;
  const int rowbase = blockIdx.x * 128 + wave * 16;
  const int m       = lane & 15;             // row within 16-row group
  const int khalf   = (lane >> 4) * 2;       // K offset inside each K=4 chunk

  const float* prow = P + ((size_t)(rowbase + m) << 14);

  v8f   c0 = {}, c1 = {};                    // two independent XDL chains
  float sq0 = 0.f, sq1 = 0.f, sq2 = 0.f, sq3 = 0.f;

  for (int k = 0; k < D; k += 8) {
    v2f a0 = __builtin_nontemporal_load((const v2f*)(prow + k + khalf));
    v2f a1 = __builtin_nontemporal_load((const v2f*)(prow + k + 4 + khalf));
    v2f b0 = *(const v2f*)(xs + k + khalf);
    v2f b1 = *(const v2f*)(xs + k + 4 + khalf);
    c0 = __builtin_amdgcn_wmma_f32_16x16x4_f32(false, a0, false, b0,
                                               (short)0, c0, false, false);
    c1 = __builtin_amdgcn_wmma_f32_16x16x4_f32(false, a1, false, b1,
                                               (short)0, c1, false, false);
    sq0 = __builtin_fmaf(a0.x, a0.x, sq0);
    sq1 = __builtin_fmaf(a0.y, a0.y, sq1);
    sq2 = __builtin_fmaf(a1.x, a1.x, sq2);
    sq3 = __builtin_fmaf(a1.y, a1.y, sq3);
  }

  // combine the two half-lane partial square norms for this row (wave32)
  float sq = (sq0 + sq1) + (sq2 + sq3);
  sq += __shfl_xor(sq, 16, 32);
  if (lane < 16) sqnorms[rowbase + m] = sq;

  // C/D layout: lanes 0-15 hold M=0..7 in c[0..7]; lanes 16-31 hold M=8..15
  // (all 16 columns identical since B was broadcast)
  if (lane == 0) {
#pragma unroll
    for (int v = 0; v < 8; ++v) dots[rowbase + v] = c0[v] + c1[v];
  }
  if (lane == 16) {
#pragma unroll
    for (int v = 0; v < 8; ++v) dots[rowbase + 8 + v] = c0[v] + c1[v];
  }
}

// ---------------------------------------------------------------------------
// Kernel 2: x-norm, cosine sims, numerically-stable softmax -> w[n]
// Single 1024-thread block; fixed-order tree reductions (deterministic).
// ---------------------------------------------------------------------------
__global__ __launch_bounds__(1024) void k_softmax(
    const float* __restrict__ x, const float* __restrict__ dots,
    const float* __restrict__ sqnorms, float* __restrict__ w)
{
  __shared__ float red[1024];
  const int t = threadIdx.x;

  // ||x||
  float s = 0.0f;
  for (int i = t; i < D; i += 1024) { float v = x[i]; s = fmaf(v, v, s); }
  red[t] = s; __syncthreads();
  for (int o = 512; o > 0; o >>= 1) {
    if (t < o) red[t] += red[t + o];
    __syncthreads();
  }
  const float xden = fmaxf(sqrtf(red[0]), EPSF);
  __syncthreads();

  // sims: 16 strided elements per thread, kept in registers
  float sim[16];
#pragma unroll
  for (int i = 0; i < 16; ++i) {
    const int n = t + i * 1024;
    const float pn = fmaxf(sqrtf(sqnorms[n]), EPSF);
    sim[i] = dots[n] / (pn * xden);
  }

  // global max
  float mx = sim[0];
#pragma unroll
  for (int i = 1; i < 16; ++i) mx = fmaxf(mx, sim[i]);
  red[t] = mx; __syncthreads();
  for (int o = 512; o > 0; o >>= 1) {
    if (t < o) red[t] = fmaxf(red[t], red[t + o]);
    __syncthreads();
  }
  const float gmax = red[0];
  __syncthreads();

  // exp + sum
  float e[16]; float es = 0.0f;
#pragma unroll
  for (int i = 0; i < 16; ++i) { e[i] = expf(sim[i] - gmax); es += e[i]; }
  red[t] = es; __syncthreads();
  for (int o = 512; o > 0; o >>= 1) {
    if (t < o) red[t] += red[t + o];
    __syncthreads();
  }
  const float inv = 1.0f / red[0];
#pragma unroll
  for (int i = 0; i < 16; ++i) w[t + i * 1024] = e[i] * inv;
}

// ---------------------------------------------------------------------------
// Kernel 3: partial[s][d] = sum_{n in split s} w[n] * P[n][d]
// grid = (16 column blocks of 1024 cols, 32 N-splits); v4f coalesced NT loads
// + global_prefetch_b8 look-ahead.
// ---------------------------------------------------------------------------
__global__ __launch_bounds__(256) void k_wsum_partial(
    const float* __restrict__ P, const float* __restrict__ w,
    float* __restrict__ partial)
{
  __shared__ float wl[ROWS_PER_SPLIT];
  const int colb  = blockIdx.x;   // 0..15
  const int split = blockIdx.y;   // 0..31
  const int n0    = split * ROWS_PER_SPLIT;

  for (int i = threadIdx.x; i < ROWS_PER_SPLIT; i += 256) wl[i] = w[n0 + i];
  __syncthreads();

  const int col0 = colb * 1024 + threadIdx.x * 4;
  const float* base = P + ((size_t)n0 << 14) + col0;

  v4f acc = {0.0f, 0.0f, 0.0f, 0.0f};
  for (int i = 0; i < ROWS_PER_SPLIT; ++i) {
    const float* p = base + ((size_t)i << 14);
    __builtin_prefetch(p + ((size_t)4 << 14), 0, 0);  // global_prefetch_b8
    const v4f v = __builtin_nontemporal_load((const v4f*)p);
    const float wv = wl[i];
    acc = __builtin_elementwise_fma((v4f){wv, wv, wv, wv}, v, acc);
  }
  *(v4f*)(partial + (size_t)split * D + col0) = acc;
}

// ---------------------------------------------------------------------------
// Kernel 4: out[d] = (1/NPAT) * sum_s partial[s][d]
// ---------------------------------------------------------------------------
__global__ __launch_bounds__(256) void k_reduce_out(
    const float* __restrict__ partial, float* __restrict__ out)
{
  const int d = blockIdx.x * 256 + threadIdx.x;
  float s = 0.0f;
#pragma unroll 4
  for (int i = 0; i < NSPLIT; ++i) s += partial[(size_t)i * D + d];
  out[d] = s * (1.0f / (float)NPAT);
}

// ---------------------------------------------------------------------------
extern "C" void kernel_launch(void* const* d_in, const int* in_sizes, int n_in,
                              void* d_out, int out_size, void* d_ws, size_t ws_size,
                              hipStream_t stream) {
  const float* x = (const float*)d_in[0];      // 16384 f32
  const float* P = (const float*)d_in[1];      // 16384*16384 f32
  float* out = (float*)d_out;                  // 16384 f32

  float* ws      = (float*)d_ws;
  float* dots    = ws;                 // [NPAT]
  float* sqn     = ws + NPAT;          // [NPAT]
  float* w       = ws + 2 * NPAT;      // [NPAT]
  float* partial = ws + 3 * NPAT;      // [NSPLIT * D] = 2 MiB

  k_dots_norms<<<NPAT / 128, 256, 0, stream>>>(P, x, dots, sqn);
  k_softmax<<<1, 1024, 0, stream>>>(x, dots, sqn, w);
  dim3 g3(16, NSPLIT);
  k_wsum_partial<<<g3, 256, 0, stream>>>(P, w, partial);
  k_reduce_out<<<D / 256, 256, 0, stream>>>(partial, out);
}